// TvChambolleDenoiser_73323681677619
// MI455X (gfx1250) — compile-verified
//
#include <hip/hip_runtime.h>
#include <math.h>

// ---------------------------------------------------------------------------
// TV-Chambolle denoiser, MI455X (gfx1250).
// 24 slices x 512x512, 10 iterations. Bandwidth-bound (~1.45GB moved vs
// ~1.3GFLOP); working set (~96MB) fits the 192MB L2. Strategy:
//   * per-iteration fused stencil kernel, tiles staged via gfx1250
//     async global->LDS copies (ASYNCcnt) -- the CDNA5 data-movement path
//   * deterministic two-level reduction for the per-slice energy Et
//   * per-slice `done` flag reproduces the reference's convergence freeze
// ---------------------------------------------------------------------------

#define NSLICE 24
#define HH 512
#define WW 512
#define NUMELI (HH * WW)
#define ROWS 4                 // rows of output per block
#define RB (HH / ROWS)         // 128 row-blocks per slice
#define TPB 256
#define TAUC 0.25f
#define EPSV 2e-4f

// GCC-style int 4-vector, as required by the async-load builtin signature
typedef int v4i __attribute__((vector_size(4 * sizeof(int))));
typedef __attribute__((address_space(1))) v4i* v4i_gptr;
typedef __attribute__((address_space(3))) v4i* v4i_lptr;

// ---- gfx1250 async global->LDS copy + wait (builtin preferred, asm fallback)
__device__ __forceinline__ void async_ld_b128(const float* g, float* l) {
#if defined(__AMDGCN__) && __has_builtin(__builtin_amdgcn_global_load_async_to_lds_b128)
  __builtin_amdgcn_global_load_async_to_lds_b128(
      (v4i_gptr)g, (v4i_lptr)l, /*imm offset*/ 0, /*cpol*/ 0);
#else
  asm volatile("global_load_async_to_lds_b128 %0, %1, off"
               :: "v"((unsigned)(unsigned long long)l), "v"(g)
               : "memory");
#endif
}

__device__ __forceinline__ void wait_async0() {
#if defined(__AMDGCN__) && __has_builtin(__builtin_amdgcn_s_wait_asynccnt)
  __builtin_amdgcn_s_wait_asynccnt(0);
#else
  asm volatile("s_wait_asynccnt 0" ::: "memory");
#endif
}

// ---------------------------------------------------------------------------
// Stencil kernel: one iteration for one 4-row band of one slice.
//   dt[y][x]  = -(p0[y][x]+p1[y][x]) + (y>0 ? p0[y-1][x]:0) + (x>0 ? p1[y][x-1]:0)
//   out       = img + dt                  (iter==0: pt==0 -> dt==0 -> out=img)
//   g0,g1     = forward diffs of out (zero-padded at far edge)
//   norm      = sqrt(g0^2+g1^2) (0 if 0)
//   pt_new    = (pt - tau*g) / (1 + (tau/w)*norm)
//   partials  = (sum dt^2, sum norm) for this band
// ---------------------------------------------------------------------------
__global__ __launch_bounds__(TPB) void tv_stencil(
    const float* __restrict__ img, const float* __restrict__ wgt,
    float* __restrict__ pt0, float* __restrict__ pt1,
    float* __restrict__ outp, float* __restrict__ partials,
    const int* __restrict__ done, int iter)
{
  const int s = blockIdx.y;
  if (done[s]) return;                       // slice frozen (uniform exit)
  const int rb  = blockIdx.x;
  const int y0  = rb * ROWS;
  const int tid = threadIdx.x;

  __shared__ float sp0[(ROWS + 2) * WW];     // pt0 rows y0-1 .. y0+ROWS
  __shared__ float sp1[(ROWS + 1) * WW];     // pt1 rows y0   .. y0+ROWS
  __shared__ float sim[(ROWS + 1) * WW];     // img rows y0   .. y0+ROWS
  __shared__ float sout[(ROWS + 1) * WW];    // out rows y0   .. y0+ROWS
  __shared__ float sred[2][TPB / 32];

  const float* imgs = img + (size_t)s * NUMELI;
  float* p0s  = pt0  + (size_t)s * NUMELI;
  float* p1s  = pt1  + (size_t)s * NUMELI;
  float* outs = outp + (size_t)s * NUMELI;
  const float w    = wgt[s / 3];
  const float tauw = TAUC / w;

  // ---- phase 1: stage tiles via async global->LDS (zero-fill out-of-range
  //      rows; iter==0 treats pt as all-zero without touching memory)
  const int NJOB = (ROWS + 2) + 2 * (ROWS + 1);          // 16 rows
  for (int c = tid; c < NJOB * (WW / 4); c += TPB) {
    const int j  = c / (WW / 4);
    const int x4 = (c % (WW / 4)) * 4;
    float* ld; const float* gs; int gy; bool ga;
    if (j < ROWS + 2) {
      gy = y0 - 1 + j;
      ld = sp0 + j * WW + x4;
      gs = p0s + (size_t)(unsigned)gy * WW + x4;
      ga = (iter > 0) && ((unsigned)gy < (unsigned)HH);
    } else if (j < 2 * ROWS + 3) {
      const int l = j - (ROWS + 2);
      gy = y0 + l;
      ld = sp1 + l * WW + x4;
      gs = p1s + (size_t)(unsigned)gy * WW + x4;
      ga = (iter > 0) && ((unsigned)gy < (unsigned)HH);
    } else {
      const int l = j - (2 * ROWS + 3);
      gy = y0 + l;
      ld = sim + l * WW + x4;
      gs = imgs + (size_t)(unsigned)gy * WW + x4;
      ga = ((unsigned)gy < (unsigned)HH);
    }
    if (ga) {
      async_ld_b128(gs, ld);
    } else {
      ld[0] = 0.f; ld[1] = 0.f; ld[2] = 0.f; ld[3] = 0.f;
    }
  }
  wait_async0();
  __syncthreads();

  // ---- phase 2: out = img + div(pt) over ROWS+1 rows (incl. bottom halo)
  for (int c = tid; c < (ROWS + 1) * WW; c += TPB) {
    const int l = c >> 9;            // WW == 512
    const int x = c & (WW - 1);
    const float p0c  = sp0[(l + 1) * WW + x];
    const float p1c  = sp1[l * WW + x];
    const float top  = sp0[l * WW + x];
    const float left = (x > 0) ? sp1[l * WW + x - 1] : 0.f;
    const float dt   = -(p0c + p1c) + top + left;
    sout[c] = sim[c] + dt;
  }
  __syncthreads();

  // ---- phase 3: gradients, norm, dual update, energy partials
  float dt2 = 0.f, nsm = 0.f;
  for (int c = tid; c < ROWS * WW; c += TPB) {
    const int l = c >> 9;
    const int x = c & (WW - 1);
    const int y = y0 + l;
    const float oc = sout[l * WW + x];
    const float g0 = (y < HH - 1) ? sout[(l + 1) * WW + x] - oc : 0.f;
    const float g1 = (x < WW - 1) ? sout[l * WW + x + 1] - oc : 0.f;
    const float ss = g0 * g0 + g1 * g1;
    const float nrm = (ss > 0.f) ? sqrtf(ss) : 0.f;
    nsm += nrm;

    const float p0c  = sp0[(l + 1) * WW + x];
    const float p1c  = sp1[l * WW + x];
    const float top  = sp0[l * WW + x];
    const float left = (x > 0) ? sp1[l * WW + x - 1] : 0.f;
    const float dt   = -(p0c + p1c) + top + left;
    dt2 += dt * dt;

    const float den = 1.0f + tauw * nrm;
    const size_t gidx = (size_t)y * WW + x;
    p0s[gidx]  = (p0c - TAUC * g0) / den;
    p1s[gidx]  = (p1c - TAUC * g1) / den;
    outs[gidx] = oc;
  }

  // ---- deterministic block reduction (wave32 shuffles + LDS)
  for (int m = 16; m > 0; m >>= 1) {
    dt2 += __shfl_xor(dt2, m, 32);
    nsm += __shfl_xor(nsm, m, 32);
  }
  const int wv = tid >> 5, ln = tid & 31;
  if (ln == 0) { sred[0][wv] = dt2; sred[1][wv] = nsm; }
  __syncthreads();
  if (tid == 0) {
    float a = 0.f, b = 0.f;
#pragma unroll
    for (int q = 0; q < TPB / 32; ++q) { a += sred[0][q]; b += sred[1][q]; }
    const size_t pi = ((size_t)s * RB + rb) * 2;
    partials[pi]     = a;
    partials[pi + 1] = b;
  }
}

// ---------------------------------------------------------------------------
// Per-slice energy finalize + convergence check (1 wave per slice,
// fixed summation order -> deterministic).
// ---------------------------------------------------------------------------
__global__ __launch_bounds__(32) void tv_reduce(
    const float* __restrict__ partials, const float* __restrict__ wgt,
    float* __restrict__ Einit, float* __restrict__ Eprev,
    int* __restrict__ done, int iter)
{
  const int s = blockIdx.x;
  if (done[s]) return;
  const int t = threadIdx.x;
  float a = 0.f, b = 0.f;
  for (int r = t; r < RB; r += 32) {
    const size_t pi = ((size_t)s * RB + r) * 2;
    a += partials[pi];
    b += partials[pi + 1];
  }
  for (int m = 16; m > 0; m >>= 1) {
    a += __shfl_xor(a, m, 32);
    b += __shfl_xor(b, m, 32);
  }
  if (t == 0) {
    const float w  = wgt[s / 3];
    const float Et = (a + w * b) / (float)NUMELI;
    if (iter == 0) {
      Einit[s] = Et;
      Eprev[s] = Et;
    } else {
      const bool conv = fabsf(Eprev[s] - Et) < EPSV * Einit[s];
      Eprev[s] = Et;
      if (conv) done[s] = 1;
    }
  }
}

__global__ void tv_init(int* __restrict__ done) {
  if (threadIdx.x < NSLICE) done[threadIdx.x] = 0;
}

// ---------------------------------------------------------------------------
extern "C" void kernel_launch(void* const* d_in, const int* in_sizes, int n_in,
                              void* d_out, int out_size, void* d_ws, size_t ws_size,
                              hipStream_t stream) {
  const float* img = (const float*)d_in[0];   // (8,3,512,512) f32
  const float* wgt = (const float*)d_in[1];   // (8,) f32
  float* outp = (float*)d_out;                // (8,3,512,512) f32

  // workspace layout (bytes):
  //   pt0      : 24 * 512*512 * 4 = 25165824
  //   pt1      : 25165824
  //   partials : 24 * 128 * 2 * 4 = 24576
  //   Einit    : 96,  Eprev : 96,  done : 96
  char* ws = (char*)d_ws;
  float* pt0      = (float*)(ws);
  float* pt1      = (float*)(ws + 25165824);
  float* partials = (float*)(ws + 2 * 25165824);
  float* Einit    = (float*)(ws + 2 * 25165824 + 24576);
  float* Eprev    = Einit + NSLICE;
  int*   done     = (int*)(Eprev + NSLICE);

  tv_init<<<1, 32, 0, stream>>>(done);
  for (int it = 0; it < 10; ++it) {
    tv_stencil<<<dim3(RB, NSLICE), TPB, 0, stream>>>(
        img, wgt, pt0, pt1, outp, partials, done, it);
    tv_reduce<<<NSLICE, 32, 0, stream>>>(partials, wgt, Einit, Eprev, done, it);
  }
}